// GConvNet_26310969655870
// MI455X (gfx1250) — compile-verified
//
#include <hip/hip_runtime.h>
#include <hip/hip_bf16.h>
#include <math.h>

#define T_WIN 10
#define N_NODES 50000
#define N_EDGES 800000
#define F_IN 128

typedef __attribute__((ext_vector_type(2))) float v2f;
typedef __attribute__((ext_vector_type(8))) float v8f;

// ---------------------------------------------------------------- degrees
__global__ void deg_kernel(const int* __restrict__ src, const int* __restrict__ dst,
                           int* __restrict__ degS, int* __restrict__ degD, int nEdges) {
    int e = blockIdx.x * blockDim.x + threadIdx.x;
    if (e >= nEdges) return;
    atomicAdd(&degS[src[e]], 1);
    atomicAdd(&degD[dst[e]], 1);
}

__global__ void norm_kernel(const int* __restrict__ degS, const int* __restrict__ degD,
                            float* __restrict__ nsrc, float* __restrict__ ndst, int nNodes) {
    int n = blockIdx.x * blockDim.x + threadIdx.x;
    if (n >= nNodes) return;
    int ds = degS[n] > 1 ? degS[n] : 1;
    int dd = degD[n] > 1 ? degD[n] : 1;
    nsrc[n] = 1.0f / sqrtf((float)ds);
    ndst[n] = 1.0f / sqrtf((float)dd);
}

// --------------------------------------------- layer1: (feat*nsrc) @ W1 (128->8)
// One wave computes one 16-row tile via V_WMMA_F32_16X16X4_F32, K=128 in 32 steps.
__global__ void gemm1_wmma(const float* __restrict__ feat, const float* __restrict__ W1,
                           const float* __restrict__ nsrc, float* __restrict__ h1a,
                           int nRows, int nNodes) {
    __shared__ float w1s[F_IN * 16];              // W1 zero-padded 8 -> 16 columns
    for (int i = threadIdx.x; i < F_IN * 16; i += blockDim.x) {
        int k = i >> 4, col = i & 15;
        w1s[i] = (col < 8) ? W1[k * 8 + col] : 0.0f;
    }
    __syncthreads();

    int wavesPerBlock = blockDim.x >> 5;
    int tile = blockIdx.x * wavesPerBlock + (threadIdx.x >> 5);
    int nTiles = nRows >> 4;
    if (tile >= nTiles) return;                   // wave-uniform exit (EXEC stays all-1)

    int lane = threadIdx.x & 31;
    int m    = lane & 15;                         // A: row / B,D: column
    int hi   = lane >> 4;                         // K-half select
    int row0 = tile << 4;
    int row  = row0 + m;
    int node = row % nNodes;
    float ns = nsrc[node];
    const float* arow = feat + (size_t)row * F_IN;

    v8f c = {0.f, 0.f, 0.f, 0.f, 0.f, 0.f, 0.f, 0.f};
    for (int kb = 0; kb < F_IN; kb += 4) {
        int k = kb + hi * 2;
        v2f a, b;
        a.x = arow[k]     * ns;
        a.y = arow[k + 1] * ns;
        b.x = w1s[k * 16 + m];
        b.y = w1s[(k + 1) * 16 + m];
        c = __builtin_amdgcn_wmma_f32_16x16x4_f32(false, a, false, b, (short)0, c, false, false);
    }
    if (m < 8) {                                  // only 8 real output columns
        int rb = row0 + hi * 8;
#pragma unroll
        for (int i = 0; i < 8; i++)
            h1a[(size_t)(rb + i) * 8 + m] = c[i];
    }
}

// ------------------------------------------------- edge scatter-sum (8 feats)
__global__ void scatter_kernel(const float* __restrict__ h, const int* __restrict__ src,
                               const int* __restrict__ dst, float* __restrict__ agg,
                               int nEdges, int nNodes) {
    int e = blockIdx.x * blockDim.x + threadIdx.x;
    if (e >= nEdges) return;
    int s = src[e], d = dst[e];
    for (int t = 0; t < T_WIN; t++) {
        const float4* hp = (const float4*)(h + ((size_t)t * nNodes + s) * 8);
        float4 v0 = hp[0], v1 = hp[1];
        float* ap = agg + ((size_t)t * nNodes + d) * 8;
        atomicAdd(ap + 0, v0.x); atomicAdd(ap + 1, v0.y);
        atomicAdd(ap + 2, v0.z); atomicAdd(ap + 3, v0.w);
        atomicAdd(ap + 4, v1.x); atomicAdd(ap + 5, v1.y);
        atomicAdd(ap + 6, v1.z); atomicAdd(ap + 7, v1.w);
    }
}

// ------------- fuse: out1 = relu(agg1*ndst + b1), then pre-scale by nsrc (in place)
__global__ void fuse1_kernel(float* __restrict__ buf, const float* __restrict__ b1,
                             const float* __restrict__ ndst, const float* __restrict__ nsrc,
                             int nRows, int nNodes) {
    int r = blockIdx.x * blockDim.x + threadIdx.x;
    if (r >= nRows) return;
    int n = r % nNodes;
    float nd = ndst[n], ns = nsrc[n];
    float* p = buf + (size_t)r * 8;
#pragma unroll
    for (int j = 0; j < 8; j++)
        p[j] = fmaxf(p[j] * nd + b1[j], 0.0f) * ns;
}

// --------------- layer2: relu((agg2*ndst) @ W2 + b2)  (8->16), K=8 -> 2 WMMAs
__global__ void gemm2_wmma(const float* __restrict__ agg2, const float* __restrict__ W2,
                           const float* __restrict__ b2, const float* __restrict__ ndst,
                           float* __restrict__ out2, int nRows, int nNodes) {
    __shared__ float w2s[8 * 16];
    if (threadIdx.x < 128) w2s[threadIdx.x] = W2[threadIdx.x];
    __syncthreads();

    int wavesPerBlock = blockDim.x >> 5;
    int tile = blockIdx.x * wavesPerBlock + (threadIdx.x >> 5);
    int nTiles = nRows >> 4;
    if (tile >= nTiles) return;

    int lane = threadIdx.x & 31;
    int m    = lane & 15;
    int hi   = lane >> 4;
    int row0 = tile << 4;
    int row  = row0 + m;
    int node = row % nNodes;
    float nd = ndst[node];
    const float* arow = agg2 + (size_t)row * 8;

    v8f c = {0.f, 0.f, 0.f, 0.f, 0.f, 0.f, 0.f, 0.f};
#pragma unroll
    for (int kb = 0; kb < 8; kb += 4) {
        int k = kb + hi * 2;
        v2f a, b;
        a.x = arow[k]     * nd;
        a.y = arow[k + 1] * nd;
        b.x = w2s[k * 16 + m];
        b.y = w2s[(k + 1) * 16 + m];
        c = __builtin_amdgcn_wmma_f32_16x16x4_f32(false, a, false, b, (short)0, c, false, false);
    }
    float bias = b2[m];
    int rb = row0 + hi * 8;
#pragma unroll
    for (int i = 0; i < 8; i++)
        out2[(size_t)(rb + i) * 16 + m] = fmaxf(c[i] + bias, 0.0f);
}

// ------------------------------- LSTM: one thread per node, weights in LDS
__global__ void lstm_kernel(const float* __restrict__ x, const float* __restrict__ Wih,
                            const float* __restrict__ Whh, const float* __restrict__ bih,
                            const float* __restrict__ bhh, float* __restrict__ hfin,
                            int nNodes) {
    __shared__ float wih[32 * 16];
    __shared__ float whh[32 * 8];
    __shared__ float bsum[32];
    for (int i = threadIdx.x; i < 32 * 16; i += blockDim.x) wih[i] = Wih[i];
    for (int i = threadIdx.x; i < 32 * 8;  i += blockDim.x) whh[i] = Whh[i];
    if (threadIdx.x < 32) bsum[threadIdx.x] = bih[threadIdx.x] + bhh[threadIdx.x];
    __syncthreads();

    int n = blockIdx.x * blockDim.x + threadIdx.x;
    if (n >= nNodes) return;

    float h[8], cc[8];
#pragma unroll
    for (int j = 0; j < 8; j++) { h[j] = 0.0f; cc[j] = 0.0f; }

    for (int t = 0; t < T_WIN; t++) {
        float xv[16];
        const float4* xp = (const float4*)(x + ((size_t)t * nNodes + n) * 16);
#pragma unroll
        for (int q = 0; q < 4; q++) {
            float4 v = xp[q];
            xv[q * 4 + 0] = v.x; xv[q * 4 + 1] = v.y;
            xv[q * 4 + 2] = v.z; xv[q * 4 + 3] = v.w;
        }
        float gates[32];
        for (int g = 0; g < 32; g++) {
            float acc = bsum[g];
#pragma unroll
            for (int k = 0; k < 16; k++) acc += xv[k] * wih[g * 16 + k];
#pragma unroll
            for (int k = 0; k < 8;  k++) acc += h[k] * whh[g * 8 + k];
            gates[g] = acc;
        }
#pragma unroll
        for (int j = 0; j < 8; j++) {
            float ig = 1.0f / (1.0f + expf(-gates[j]));
            float fg = 1.0f / (1.0f + expf(-gates[8 + j]));
            float gg = tanhf(gates[16 + j]);
            float og = 1.0f / (1.0f + expf(-gates[24 + j]));
            cc[j] = fg * cc[j] + ig * gg;
            h[j]  = og * tanhf(cc[j]);
        }
    }
#pragma unroll
    for (int j = 0; j < 8; j++) hfin[(size_t)n * 8 + j] = h[j];
}

// ------------------------------------------------------- max pool (2 stages)
__global__ void pool1_kernel(const float* __restrict__ hfin, float* __restrict__ partial,
                             int nNodes) {
    __shared__ float red[256 * 8];
    float m[8];
#pragma unroll
    for (int j = 0; j < 8; j++) m[j] = -1e30f;
    int stride = gridDim.x * blockDim.x;
    for (int n = blockIdx.x * blockDim.x + threadIdx.x; n < nNodes; n += stride) {
#pragma unroll
        for (int j = 0; j < 8; j++) m[j] = fmaxf(m[j], hfin[(size_t)n * 8 + j]);
    }
#pragma unroll
    for (int j = 0; j < 8; j++) red[threadIdx.x * 8 + j] = m[j];
    __syncthreads();
    for (int s = blockDim.x >> 1; s > 0; s >>= 1) {
        if (threadIdx.x < s)
#pragma unroll
            for (int j = 0; j < 8; j++)
                red[threadIdx.x * 8 + j] =
                    fmaxf(red[threadIdx.x * 8 + j], red[(threadIdx.x + s) * 8 + j]);
        __syncthreads();
    }
    if (threadIdx.x == 0)
#pragma unroll
        for (int j = 0; j < 8; j++) partial[blockIdx.x * 8 + j] = red[j];
}

__global__ void pool2_kernel(const float* __restrict__ partial, int nPart,
                             const float* __restrict__ Wo, const float* __restrict__ bo,
                             float* __restrict__ out) {
    __shared__ float pooled[8];
    if (threadIdx.x < 8) {
        float m = -1e30f;
        for (int b = 0; b < nPart; b++) m = fmaxf(m, partial[b * 8 + threadIdx.x]);
        pooled[threadIdx.x] = m;
    }
    __syncthreads();
    if (threadIdx.x < 4) {
        float acc = bo[threadIdx.x];
#pragma unroll
        for (int k = 0; k < 8; k++) acc += pooled[k] * Wo[k * 4 + threadIdx.x];
        out[threadIdx.x] = 1.0f / (1.0f + expf(-acc));
    }
}

// ------------------------------------------------------------------ launch
extern "C" void kernel_launch(void* const* d_in, const int* in_sizes, int n_in,
                              void* d_out, int out_size, void* d_ws, size_t ws_size,
                              hipStream_t stream) {
    const float* feat = (const float*)d_in[0];
    const int*   src  = (const int*)d_in[1];
    const int*   dst  = (const int*)d_in[2];
    const float* W1   = (const float*)d_in[3];
    const float* b1   = (const float*)d_in[4];
    const float* W2   = (const float*)d_in[5];
    const float* b2   = (const float*)d_in[6];
    const float* Wih  = (const float*)d_in[7];
    const float* Whh  = (const float*)d_in[8];
    const float* bih  = (const float*)d_in[9];
    const float* bhh  = (const float*)d_in[10];
    const float* Wo   = (const float*)d_in[11];
    const float* bo   = (const float*)d_in[12];
    float* out = (float*)d_out;

    const int N = N_NODES, E = N_EDGES;
    const int nRows = T_WIN * N;                       // 500000 (divisible by 16)

    // workspace layout
    char* p = (char*)d_ws;
    int*   degS = (int*)p;                 p += (size_t)N * 4;
    int*   degD = (int*)p;                 p += (size_t)N * 4;
    float* nsrc = (float*)p;               p += (size_t)N * 4;
    float* ndst = (float*)p;               p += (size_t)N * 4;
    float* bufA = (float*)p;               p += (size_t)nRows * 8 * 4;   // h1a / agg2
    float* bufB = (float*)p;               p += (size_t)nRows * 8 * 4;   // agg1 / h2pre
    float* bufC = (float*)p;               p += (size_t)nRows * 16 * 4;  // out2 (LSTM in)
    float* hfin = (float*)p;               p += (size_t)N * 8 * 4;
    float* partial = (float*)p;            p += (size_t)128 * 8 * 4;

    const int POOL_BLOCKS = 128;

    // degrees + norms
    hipMemsetAsync(degS, 0, (size_t)2 * N * 4, stream);   // degS+degD contiguous
    deg_kernel<<<(E + 255) / 256, 256, 0, stream>>>(src, dst, degS, degD, E);
    norm_kernel<<<(N + 255) / 256, 256, 0, stream>>>(degS, degD, nsrc, ndst, N);

    // layer 1: GEMM (WMMA f32) -> scatter -> fused norm/bias/relu/prescale
    int nTiles = nRows / 16;                              // 31250
    int wavesPerBlock = 256 / 32;
    int gemmBlocks = (nTiles + wavesPerBlock - 1) / wavesPerBlock;
    gemm1_wmma<<<gemmBlocks, 256, 0, stream>>>(feat, W1, nsrc, bufA, nRows, N);

    hipMemsetAsync(bufB, 0, (size_t)nRows * 8 * 4, stream);
    scatter_kernel<<<(E + 255) / 256, 256, 0, stream>>>(bufA, src, dst, bufB, E, N);
    fuse1_kernel<<<(nRows + 255) / 256, 256, 0, stream>>>(bufB, b1, ndst, nsrc, nRows, N);

    // layer 2: scatter -> GEMM (WMMA f32) with ndst fold + bias + relu
    hipMemsetAsync(bufA, 0, (size_t)nRows * 8 * 4, stream);
    scatter_kernel<<<(E + 255) / 256, 256, 0, stream>>>(bufB, src, dst, bufA, E, N);
    gemm2_wmma<<<gemmBlocks, 256, 0, stream>>>(bufA, W2, b2, ndst, bufC, nRows, N);

    // LSTM + max pool + head
    lstm_kernel<<<(N + 255) / 256, 256, 0, stream>>>(bufC, Wih, Whh, bih, bhh, hfin, N);
    pool1_kernel<<<POOL_BLOCKS, 256, 0, stream>>>(hfin, partial, N);
    pool2_kernel<<<1, 64, 0, stream>>>(partial, POOL_BLOCKS, Wo, bo, out);
}